// GraphConvolution_29557964931231
// MI455X (gfx1250) — compile-verified
//
#include <hip/hip_runtime.h>

#define FD 64   // in/out feature dim

typedef __attribute__((ext_vector_type(2))) float v2f;
typedef __attribute__((ext_vector_type(8))) float v8f;

// ---------------- Kernel 1: support = s * h0  (fold alpha*h0 into accumulator) ----------------
__global__ void gcn2_init_support(const float* __restrict__ h0,
                                  const float* __restrict__ s_p,
                                  float* __restrict__ support,
                                  int total4) {
  int i = blockIdx.x * blockDim.x + threadIdx.x;
  if (i >= total4) return;
  float s = s_p[0];
  float4 v = ((const float4*)h0)[i];
  float4 o = make_float4(s * v.x, s * v.y, s * v.z, s * v.w);
  ((float4*)support)[i] = o;
}

// ---------------- Kernel 2: scatter-atomic SpMM --------------------------------------------
// support[row[e]][:] += (1-s) * ev[e] * x[col[e]][:]
// One thread per (edge, 4-feature chunk): 16 threads (half-wave) cover one edge's 64 features,
// so the x-row gather and the atomic targets are contiguous 256B regions (L2-resident:
// x = 12.8 MB, support = 12.8 MB, both << 192 MB L2).
__global__ void gcn2_spmm_scatter(const float* __restrict__ x,
                                  const float* __restrict__ ev,
                                  const int* __restrict__ row,
                                  const int* __restrict__ col,
                                  const float* __restrict__ s_p,
                                  float* __restrict__ support,
                                  int E_) {
  int t = blockIdx.x * blockDim.x + threadIdx.x;
  int e = t >> 4;
  if (e >= E_) return;
  int f = (t & 15) << 2;                 // feature offset 0..60
  float a = (1.0f - s_p[0]) * ev[e];
  int c = col[e];
  int r = row[e];
  float4 xv = *(const float4*)(x + c * FD + f);
  float* dst = support + (size_t)r * FD + f;
  __hip_atomic_fetch_add(dst + 0, a * xv.x, __ATOMIC_RELAXED, __HIP_MEMORY_SCOPE_AGENT);
  __hip_atomic_fetch_add(dst + 1, a * xv.y, __ATOMIC_RELAXED, __HIP_MEMORY_SCOPE_AGENT);
  __hip_atomic_fetch_add(dst + 2, a * xv.z, __ATOMIC_RELAXED, __HIP_MEMORY_SCOPE_AGENT);
  __hip_atomic_fetch_add(dst + 3, a * xv.w, __ATOMIC_RELAXED, __HIP_MEMORY_SCOPE_AGENT);
}

// ---------------- Kernel 3: out = theta*(support@W) + (1-theta)*support  (WMMA fp32) -------
// One wave32 per 16-row slab of nodes. C/D layout: elem(M = r + 8*(lane>>4), N = lane&15).
// A(16x4 f32): lane L holds row M=L&15; VGPR0/1 = K = k+2*(L>>4), k+2*(L>>4)+1.
// B(4x16):    lane L holds col N=L&15; VGPR0/1 = rows K = k+2*(L>>4), k+2*(L>>4)+1.
//
// W is staged in LDS in K-pair-interleaved layout so each B fragment is ONE aligned
// ds_load_b64 landing directly in an even-aligned VGPR pair (no repack movs):
//   Wp[(k/2)*64 + n] = { W[k][n], W[k+1][n] }
__global__ void __launch_bounds__(128)
gcn2_gemm_wmma(const float* __restrict__ support,
               const float* __restrict__ W,
               const float* __restrict__ lamda,
               const int* __restrict__ l_p,
               float* __restrict__ out,
               int N_) {
  __shared__ float2 Wp[(FD / 2) * FD];   // 16 KB: K-pair-interleaved weight matrix
  // Cooperative pack: consecutive tid -> consecutive n => coalesced global reads.
  for (int idx = threadIdx.x; idx < (FD / 2) * FD; idx += blockDim.x) {
    int t = idx >> 6;                    // K pair index: rows 2t, 2t+1
    int n = idx & 63;
    float2 p;
    p.x = W[(2 * t    ) * FD + n];
    p.y = W[(2 * t + 1) * FD + n];
    Wp[idx] = p;
  }
  __syncthreads();

  const float theta = lamda[0] / (float)l_p[0];
  const float omt   = 1.0f - theta;

  const int wavesPerBlock = blockDim.x >> 5;
  const int wave = blockIdx.x * wavesPerBlock + (threadIdx.x >> 5);
  const int row0 = wave * 16;
  if (row0 >= N_) return;                // wave-uniform: EXEC stays all-ones for WMMA

  const int lane = threadIdx.x & 31;
  const int half = lane >> 4;            // 0: lanes 0-15, 1: lanes 16-31
  const int m    = lane & 15;

  v8f acc0 = {}, acc1 = {}, acc2 = {}, acc3 = {};

  for (int k = 0; k < FD; k += 4) {
    // A fragment: rows of `support`, K-pair per half-wave (one global_load_b64)
    const float* ap = support + (size_t)(row0 + m) * FD + k + 2 * half;
    v2f a;
    a.x = ap[0];
    a.y = ap[1];

    // B fragments: one ds_load_b64 each, contiguous K-pair at the right lane's column
    const int bbase = ((k >> 1) + half) * FD + m;   // index into Wp (float2 units)
    float2 p0 = Wp[bbase +  0];
    float2 p1 = Wp[bbase + 16];
    float2 p2 = Wp[bbase + 32];
    float2 p3 = Wp[bbase + 48];
    v2f b0; b0.x = p0.x; b0.y = p0.y;
    v2f b1; b1.x = p1.x; b1.y = p1.y;
    v2f b2; b2.x = p2.x; b2.y = p2.y;
    v2f b3; b3.x = p3.x; b3.y = p3.y;

    acc0 = __builtin_amdgcn_wmma_f32_16x16x4_f32(false, a, false, b0, (short)0, acc0, false, false);
    acc1 = __builtin_amdgcn_wmma_f32_16x16x4_f32(false, a, false, b1, (short)0, acc1, false, false);
    acc2 = __builtin_amdgcn_wmma_f32_16x16x4_f32(false, a, false, b2, (short)0, acc2, false, false);
    acc3 = __builtin_amdgcn_wmma_f32_16x16x4_f32(false, a, false, b3, (short)0, acc3, false, false);
  }

  // Epilogue: out = theta*G + (1-theta)*support, coalesced 16-float rows per half-wave.
  #pragma unroll
  for (int r = 0; r < 8; ++r) {
    int mr = row0 + r + 8 * half;
    if (mr >= N_) continue;
    const float* sp = support + (size_t)mr * FD;
    float*       op = out     + (size_t)mr * FD;
    op[ 0 + m] = theta * acc0[r] + omt * sp[ 0 + m];
    op[16 + m] = theta * acc1[r] + omt * sp[16 + m];
    op[32 + m] = theta * acc2[r] + omt * sp[32 + m];
    op[48 + m] = theta * acc3[r] + omt * sp[48 + m];
  }
}

extern "C" void kernel_launch(void* const* d_in, const int* in_sizes, int n_in,
                              void* d_out, int out_size, void* d_ws, size_t ws_size,
                              hipStream_t stream) {
  const float* x       = (const float*)d_in[0];
  const float* h0      = (const float*)d_in[1];
  const float* ev      = (const float*)d_in[2];
  const float* W       = (const float*)d_in[3];
  const int*   erow    = (const int*)d_in[4];
  const int*   ecol    = (const int*)d_in[5];
  const float* lamda   = (const float*)d_in[6];
  const float* s_p     = (const float*)d_in[7];
  const int*   l_p     = (const int*)d_in[8];
  float*       out     = (float*)d_out;

  const int N_ = in_sizes[1] / FD;   // h0 is [N, 64]
  const int E_ = in_sizes[2];        // edge_vals is [E]

  float* support = (float*)d_ws;     // N * 64 floats (12.8 MB)

  // 1) support = s * h0
  {
    int total4 = (N_ * FD) / 4;
    int blk = 256;
    int grd = (total4 + blk - 1) / blk;
    gcn2_init_support<<<grd, blk, 0, stream>>>(h0, s_p, support, total4);
  }
  // 2) support += (1-s) * A @ x   (scatter atomics, L2-resident)
  {
    long long threads = (long long)E_ * 16;
    int blk = 256;
    int grd = (int)((threads + blk - 1) / blk);
    gcn2_spmm_scatter<<<grd, blk, 0, stream>>>(x, ev, erow, ecol, s_p, support, E_);
  }
  // 3) out = theta*(support @ W) + (1-theta)*support   (WMMA fp32)
  {
    int slabs = (N_ + 15) / 16;
    int blk = 128;                       // 4 waves/block
    int wavesPerBlock = blk / 32;
    int grd = (slabs + wavesPerBlock - 1) / wavesPerBlock;
    gcn2_gemm_wmma<<<grd, blk, 0, stream>>>(support, W, lamda, l_p, out, N_);
  }
}